// NeighborEncoder_25494925869601
// MI455X (gfx1250) — compile-verified
//
#include <hip/hip_runtime.h>
#include <math.h>

#define HIDDEN 128
#define HEADS 4

typedef float v2f __attribute__((ext_vector_type(2)));
typedef float v8f __attribute__((ext_vector_type(8)));

__device__ __forceinline__ float elu1(float x) { return x > 0.f ? x : (expf(x) - 1.f); }
// monotonic float<->int mapping so we can use integer atomicMax for segment max
__device__ __forceinline__ int f2o(float f) { int i = __float_as_int(f); return i >= 0 ? i : (i ^ 0x7FFFFFFF); }
__device__ __forceinline__ float o2f(int i) { return __int_as_float(i >= 0 ? i : (i ^ 0x7FFFFFFF)); }

// ---------------------------------------------------------------------------
// Out[r, 0:128] = act( A[r, 0:128] @ W[128,128] + bias )
// One wave computes a 16-row x 128-col tile with v_wmma_f32_16x16x4_f32.
// W is pre-swizzled in LDS into the exact per-lane B-fragment layout so each
// fragment is one aligned 64-bit LDS read (pairs combine to ds_load_2addr_b64,
// lane*8B -> perfect 64-bank spread, no conflicts, no repack movs).
// All 8 B fragments of a K-step are fetched into registers BEFORE the WMMAs
// so the 4 ds_load_2addr_b64 issue back-to-back behind one dscnt wait.
// 8 acc tiles = 64 VGPRs -> no spills.
// 8 waves / block -> 128 rows / block.  LDS: 64KB W + 512B bias << 320KB/WGP.
// ---------------------------------------------------------------------------
__global__ void __launch_bounds__(256, 1)
node_gemm_kernel(const float* __restrict__ A, const float* __restrict__ W,
                 const float* __restrict__ bias, float* __restrict__ Out,
                 int n_rows, int apply_elu)
{
    extern __shared__ float lds[];               // [32 kst][8 nt][32 lane][2] + [128] bias
    float* ldsW = lds;
    float* ldsB = lds + HIDDEN * HIDDEN;

    const int tid = threadIdx.x;
    // Pre-swizzle: fragment pair f=(kst,nt,ln): B-layout (ISA 7.12.2) is
    // lanes 0-15 -> K=kb,kb+1 at N=ln&15 ; lanes 16-31 -> K=kb+2,kb+3.
    for (int f = tid; f < (HIDDEN * HIDDEN) / 2; f += blockDim.x) {
        const int kst  = f >> 8;                 // K-step 0..31 (K = 4*kst)
        const int nt   = (f >> 5) & 7;           // N-tile 0..7
        const int ln   = f & 31;                 // lane
        const int ncol = (nt << 4) + (ln & 15);
        const int kb   = (kst << 2) + ((ln >> 4) << 1);
        ldsW[2 * f + 0] = W[kb * HIDDEN + ncol];
        ldsW[2 * f + 1] = W[(kb + 1) * HIDDEN + ncol];
    }
    if (tid < HIDDEN) ldsB[tid] = bias[tid];
    __syncthreads();

    const int wave = tid >> 5;
    const int lane = tid & 31;
    const int row0 = (blockIdx.x * (blockDim.x >> 5) + wave) << 4;   // 16 rows / wave
    if (row0 >= n_rows) return;

    const int mrow  = lane & 15;      // M-row (A) / N-col (B,C) within tile
    const int khalf = lane >> 4;      // 0: K=k,k+1  1: K=k+2,k+3

    const float* arow = A + (size_t)min(row0 + mrow, n_rows - 1) * HIDDEN;

    v8f acc[8] = {};                  // 8 N-tiles of 16 cols -> full 128 cols

    const float* frag = ldsW + (lane << 1);
    for (int kst = 0; kst < 32; ++kst) {
        const int kb = (kst << 2) + (khalf << 1);
        v2f a;
        a.x = arow[kb];
        a.y = arow[kb + 1];
        v2f b[8];
#pragma unroll
        for (int nt = 0; nt < 8; ++nt)
            b[nt] = *(const v2f*)(frag + (((kst << 3) + nt) << 6));
#pragma unroll
        for (int nt = 0; nt < 8; ++nt)
            acc[nt] = __builtin_amdgcn_wmma_f32_16x16x4_f32(
                false, a, false, b[nt], (short)0, acc[nt], false, false);
    }

    // C/D layout: VGPR i -> row (i + khalf*8), col = nt*16 + (lane&15)
#pragma unroll
    for (int nt = 0; nt < 8; ++nt) {
        const int col = (nt << 4) + mrow;
        const float bv = ldsB[col];
#pragma unroll
        for (int i = 0; i < 8; ++i) {
            const int row = row0 + i + khalf * 8;
            if (row < n_rows) {
                float v = acc[nt][i] + bv;
                if (apply_elu) v = elu1(v);
                Out[(size_t)row * HIDDEN + col] = v;
            }
        }
    }
}

// ---------------------------------------------------------------------------
// Workspace init: agg=0, softmax sums=0, segment max = ordered(-inf), mean=0
// ---------------------------------------------------------------------------
__global__ void init_ws_kernel(float* __restrict__ agg, float* __restrict__ ssum,
                               int* __restrict__ mmax, float* __restrict__ msum, int n)
{
    const long long i = (long long)blockIdx.x * blockDim.x + threadIdx.x;
    if (i < (long long)n * HIDDEN) agg[i] = 0.f;
    if (i < (long long)n * HEADS) { ssum[i] = 0.f; mmax[i] = f2o(-INFINITY); }
    if (i < 2) msum[i] = 0.f;
}

// ---------------------------------------------------------------------------
// Column sums of edge_attr [E,2] (for the self-loop 'mean' fill value)
// ---------------------------------------------------------------------------
__global__ void edge_attr_sum_kernel(const float* __restrict__ ea, int E, float* __restrict__ msum)
{
    const int stride = gridDim.x * blockDim.x;
    float s0 = 0.f, s1 = 0.f;
    for (long long e = (long long)blockIdx.x * blockDim.x + threadIdx.x; e < E; e += stride) {
        s0 += ea[2 * e];
        s1 += ea[2 * e + 1];
    }
    for (int off = 16; off; off >>= 1) {
        s0 += __shfl_down(s0, off, 32);
        s1 += __shfl_down(s1, off, 32);
    }
    if ((threadIdx.x & 31) == 0) {
        unsafeAtomicAdd(&msum[0], s0);
        unsafeAtomicAdd(&msum[1], s1);
    }
}

// ---------------------------------------------------------------------------
// Per-edge attention logits + segment max.  One wave32 per edge; lane owns
// 4 contiguous channels (float4), per-head reduce across 8-lane groups.
// ---------------------------------------------------------------------------
__global__ void edge_alpha_kernel(const int* __restrict__ ei, const float* __restrict__ ea,
                                  const float* __restrict__ msum, const float* __restrict__ w_e,
                                  const float* __restrict__ att, const float* __restrict__ xl,
                                  const float* __restrict__ xr, float* __restrict__ alpha,
                                  int* __restrict__ mmax, int E, int n)
{
    const int lane = threadIdx.x & 31;
    const long long e = (long long)blockIdx.x * (blockDim.x >> 5) + (threadIdx.x >> 5);
    if (e >= (long long)E + n) return;

    int s, d; float ea0, ea1;
    if (e < E) {
        s = ei[e]; d = ei[(size_t)E + e];
        ea0 = ea[2 * e]; ea1 = ea[2 * e + 1];
    } else {                                   // self loop, edge_attr = mean
        s = d = (int)(e - E);
        const float inv = 1.f / (float)E;
        ea0 = msum[0] * inv; ea1 = msum[1] * inv;
    }

    const int c0 = lane << 2;                  // channels c0..c0+3, head = lane>>3
    const float4 xlv = *(const float4*)(xl + (size_t)s * HIDDEN + c0);
    const float4 xrv = *(const float4*)(xr + (size_t)d * HIDDEN + c0);
    const float4 we0 = *(const float4*)(w_e + c0);
    const float4 we1 = *(const float4*)(w_e + HIDDEN + c0);
    const float4 atv = *(const float4*)(att + c0);   // att[h][c] flat == channel index

    float p = 0.f, z;
    z = xlv.x + xrv.x + ea0 * we0.x + ea1 * we1.x; z = z > 0.f ? z : 0.2f * z; p += z * atv.x;
    z = xlv.y + xrv.y + ea0 * we0.y + ea1 * we1.y; z = z > 0.f ? z : 0.2f * z; p += z * atv.y;
    z = xlv.z + xrv.z + ea0 * we0.z + ea1 * we1.z; z = z > 0.f ? z : 0.2f * z; p += z * atv.z;
    z = xlv.w + xrv.w + ea0 * we0.w + ea1 * we1.w; z = z > 0.f ? z : 0.2f * z; p += z * atv.w;

    p += __shfl_xor(p, 1, 32);                 // reduce the 8 lanes of one head
    p += __shfl_xor(p, 2, 32);
    p += __shfl_xor(p, 4, 32);

    if ((lane & 7) == 0) {
        const int h = lane >> 3;
        alpha[e * HEADS + h] = p;
        atomicMax(&mmax[(size_t)d * HEADS + h], f2o(p));
    }
}

// ---------------------------------------------------------------------------
// ex = exp(alpha - segmax[dst]); segment-sum the exps.  1 thread / (edge,head)
// ---------------------------------------------------------------------------
__global__ void edge_expsum_kernel(const int* __restrict__ ei, float* __restrict__ alpha,
                                   const int* __restrict__ mmax, float* __restrict__ ssum,
                                   int E, int n)
{
    const long long i = (long long)blockIdx.x * blockDim.x + threadIdx.x;
    if (i >= (long long)(E + n) * HEADS) return;
    const long long e = i >> 2;
    const int h = (int)(i & 3);
    const int d = (e < E) ? ei[(size_t)E + e] : (int)(e - E);
    const float m = o2f(mmax[(size_t)d * HEADS + h]);
    const float ex = expf(alpha[i] - m);
    alpha[i] = ex;
    unsafeAtomicAdd(&ssum[(size_t)d * HEADS + h], ex);
}

// ---------------------------------------------------------------------------
// agg[dst] += x_l[src] * softmax weight.  One wave per edge, float4 per lane,
// native f32 global atomics.
// ---------------------------------------------------------------------------
__global__ void edge_aggregate_kernel(const int* __restrict__ ei, const float* __restrict__ alpha,
                                      const float* __restrict__ ssum, const float* __restrict__ xl,
                                      float* __restrict__ agg, int E, int n)
{
    const int lane = threadIdx.x & 31;
    const long long e = (long long)blockIdx.x * (blockDim.x >> 5) + (threadIdx.x >> 5);
    if (e >= (long long)E + n) return;

    int s, d;
    if (e < E) { s = ei[e]; d = ei[(size_t)E + e]; }
    else       { s = d = (int)(e - E); }

    const int h = lane >> 3;
    const float a = alpha[e * HEADS + h] / ssum[(size_t)d * HEADS + h];

    const int c0 = lane << 2;
    const float4 xlv = *(const float4*)(xl + (size_t)s * HIDDEN + c0);
    float* dstp = agg + (size_t)d * HIDDEN + c0;
    unsafeAtomicAdd(dstp + 0, xlv.x * a);
    unsafeAtomicAdd(dstp + 1, xlv.y * a);
    unsafeAtomicAdd(dstp + 2, xlv.z * a);
    unsafeAtomicAdd(dstp + 3, xlv.w * a);
}

// ---------------------------------------------------------------------------
// h2 = elu(agg + bias_conv)
// ---------------------------------------------------------------------------
__global__ void bias_elu_kernel(const float* __restrict__ agg, const float* __restrict__ bias,
                                float* __restrict__ h2, int n)
{
    const long long i = (long long)blockIdx.x * blockDim.x + threadIdx.x;
    if (i >= (long long)n * HIDDEN) return;
    h2[i] = elu1(agg[i] + bias[i & (HIDDEN - 1)]);
}

// ---------------------------------------------------------------------------
extern "C" void kernel_launch(void* const* d_in, const int* in_sizes, int n_in,
                              void* d_out, int out_size, void* d_ws, size_t ws_size,
                              hipStream_t stream)
{
    const float* x    = (const float*)d_in[0];
    const int*   ei   = (const int*)  d_in[1];   // [2,E]: row0 = src, row1 = dst
    const float* ea   = (const float*)d_in[2];
    const float* w_in = (const float*)d_in[3];
    const float* b_in = (const float*)d_in[4];
    const float* w_l  = (const float*)d_in[5];
    const float* b_l  = (const float*)d_in[6];
    const float* w_r  = (const float*)d_in[7];
    const float* b_r  = (const float*)d_in[8];
    const float* w_e  = (const float*)d_in[9];
    const float* att  = (const float*)d_in[10];
    const float* b_cv = (const float*)d_in[11];
    const float* w_o  = (const float*)d_in[12];
    const float* b_o  = (const float*)d_in[13];
    float* out = (float*)d_out;

    const int n  = in_sizes[0] / HIDDEN;
    const int E  = in_sizes[2] / 2;
    const long long E2 = (long long)E + n;

    // ---- workspace carve-out (256B-aligned) ----
    char* ws = (char*)d_ws;
    size_t off = 0;
    auto carve = [&](size_t bytes) -> char* {
        char* p = ws + off;
        off += (bytes + 255) & ~(size_t)255;
        return p;
    };
    float* h     = (float*)carve((size_t)n * HIDDEN * 4);   // h, later reused for h2
    float* xl    = (float*)carve((size_t)n * HIDDEN * 4);
    float* xr    = (float*)carve((size_t)n * HIDDEN * 4);
    float* agg   = (float*)carve((size_t)n * HIDDEN * 4);
    float* alpha = (float*)carve((size_t)E2 * HEADS * 4);
    float* ssum  = (float*)carve((size_t)n * HEADS * 4);
    int*   mmax  = (int*)  carve((size_t)n * HEADS * 4);
    float* msum  = (float*)carve(2 * 4);

    const int T = 256;

    // 1. init workspace
    {
        const long long tot = (long long)n * HIDDEN;
        init_ws_kernel<<<(int)((tot + T - 1) / T), T, 0, stream>>>(agg, ssum, mmax, msum, n);
    }
    // 2. edge_attr column sums (for self-loop mean)
    edge_attr_sum_kernel<<<512, T, 0, stream>>>(ea, E, msum);

    // 3-5. node GEMMs (WMMA): h = elu(x@W_in+b); x_l = h@W_l+b; x_r = h@W_r+b
    const int lds_bytes  = (HIDDEN * HIDDEN + HIDDEN) * 4;
    const int gemm_grid  = (n + 127) / 128;                  // 8 waves * 16 rows / block
    node_gemm_kernel<<<gemm_grid, T, lds_bytes, stream>>>(x, w_in, b_in, h, n, 1);
    node_gemm_kernel<<<gemm_grid, T, lds_bytes, stream>>>(h, w_l, b_l, xl, n, 0);
    node_gemm_kernel<<<gemm_grid, T, lds_bytes, stream>>>(h, w_r, b_r, xr, n, 0);

    // 6. attention logits + segment max (wave per edge)
    const int egrid = (int)((E2 + 7) / 8);
    edge_alpha_kernel<<<egrid, T, 0, stream>>>(ei, ea, msum, w_e, att, xl, xr, alpha, mmax, E, n);

    // 7. exp + segment sum
    {
        const long long tot = E2 * HEADS;
        edge_expsum_kernel<<<(int)((tot + T - 1) / T), T, 0, stream>>>(ei, alpha, mmax, ssum, E, n);
    }
    // 8. weighted scatter-aggregate
    edge_aggregate_kernel<<<egrid, T, 0, stream>>>(ei, alpha, ssum, xl, agg, E, n);

    // 9. h2 = elu(agg + bias_conv)  (reuse h buffer)
    {
        const long long tot = (long long)n * HIDDEN;
        bias_elu_kernel<<<(int)((tot + T - 1) / T), T, 0, stream>>>(agg, b_cv, h, n);
    }
    // 10. output projection (WMMA)
    node_gemm_kernel<<<gemm_grid, T, lds_bytes, stream>>>(h, w_o, b_o, out, n, 0);
}